// CubicInterpolator_65000035058092
// MI455X (gfx1250) — compile-verified
//
#include <hip/hip_runtime.h>

typedef __attribute__((ext_vector_type(2))) float v2f;
typedef __attribute__((ext_vector_type(8))) float v8f;

#define DX 112
#define DY 224
#define DZ 160
#define SY DZ              // 160
#define SX (DY * DZ)       // 35840
#define SC (DX * DY * DZ)  // 4014080
#define NCH 8

// out[v][c] = sum_{x,y,z} wx[x]*wy[y]*wz[z] * vol[c][ix+x][iy+y][iz+z]
// Weights w_axis = [1,u,u^2,u^3] @ (0.5*KeysKernel)  -- computed via
// V_WMMA_F32_16X16X4_F32: A[m][k] = K[k][m&3] (constant), B[k][n] = u_n^k,
// D[m][n] = w[vertex n][tap m&3]  -> each lane ends with its vertex's 4 taps
// in D VGPRs 0..3, no shuffles needed.
__global__ __launch_bounds__(256) void cubic_interp_wmma(
    const float* __restrict__ vert,
    const float* __restrict__ vol,
    float* __restrict__ out,
    int V)
{
    const int lane = threadIdx.x & 31;
    const int warp = threadIdx.x >> 5;
    const int grp  = blockIdx.x * 8 + warp;   // one 16-vertex group per wave
    const int vbase = grp * 16;
    if (vbase >= V) return;                   // wave-uniform: EXEC stays full

    const int half = lane >> 4;               // 0: K=0,1 / ch 0-3 ; 1: K=2,3 / ch 4-7
    const int vsel = lane & 15;               // vertex within group == WMMA column N
    const int t    = lane & 3;                // tap index == M & 3

    // ---- A matrix (loop-invariant): A[m][k] = Kmat[k][t], Kmat = 0.5*Keys ----
    // Kmat rows: k0=[0,1,0,0] k1=[-.5,0,.5,0] k2=[1,-2.5,2,-.5] k3=[-.5,1.5,-1.5,.5]
    const float r0 = (t == 1) ? 1.0f : 0.0f;
    const float r1 = (t == 0) ? -0.5f : ((t == 2) ? 0.5f : 0.0f);
    const float r2 = (t == 0) ? 1.0f : ((t == 1) ? -2.5f : ((t == 2) ? 2.0f : -0.5f));
    const float r3 = (t == 0) ? -0.5f : ((t == 1) ? 1.5f : ((t == 2) ? -1.5f : 0.5f));
    v2f A;
    A.x = half ? r2 : r0;   // this lane supplies K = 2*half
    A.y = half ? r3 : r1;   // and K = 2*half + 1

    // ---- per-vertex coordinate -> u powers (lanes j and j+16 share vertex j) ----
    int v = vbase + vsel;
    if (v >= V) v = V - 1;   // clamp for loads; stores guarded below

    float px = vert[(size_t)v * 3 + 0];
    float py = vert[(size_t)v * 3 + 1];
    float pz = vert[(size_t)v * 3 + 2];

    px = fminf(fmaxf(px, 1.00001f), (float)(DX - 2) - 1e-5f);
    py = fminf(fmaxf(py, 1.00001f), (float)(DY - 2) - 1e-5f);
    pz = fminf(fmaxf(pz, 1.00001f), (float)(DZ - 2) - 1e-5f);

    const int ixi = (int)px;   // positive -> trunc == floor
    const int iyi = (int)py;
    const int izi = (int)pz;
    const float ux = px - (float)ixi;
    const float uy = py - (float)iyi;
    const float uz = pz - (float)izi;

    // ---- B matrices: B[k][n] = u_n^k ; lanes<16 give {1,u}, lanes>=16 {u^2,u^3} ----
    v2f Bx, By, Bz;
    {
        const float x2 = ux * ux;
        const float y2 = uy * uy;
        const float z2 = uz * uz;
        Bx.x = half ? x2 : 1.0f;  Bx.y = half ? x2 * ux : ux;
        By.x = half ? y2 : 1.0f;  By.y = half ? y2 * uy : uy;
        Bz.x = half ? z2 : 1.0f;  Bz.y = half ? z2 * uz : uz;
    }

    const v8f zero = {0.f, 0.f, 0.f, 0.f, 0.f, 0.f, 0.f, 0.f};
    // D[r] (r=0..3) = weight tap r for vertex vsel, in every lane
    v8f wx = __builtin_amdgcn_wmma_f32_16x16x4_f32(false, A, false, Bx, (short)0, zero, false, false);
    v8f wy = __builtin_amdgcn_wmma_f32_16x16x4_f32(false, A, false, By, (short)0, zero, false, false);
    v8f wz = __builtin_amdgcn_wmma_f32_16x16x4_f32(false, A, false, Bz, (short)0, zero, false, false);

    const float wz0 = wz[0], wz1 = wz[1], wz2 = wz[2], wz3 = wz[3];

    // ---- gather + reduce: this lane handles channels half*4 .. half*4+3 ----
    const float* base = vol + (size_t)(half * 4) * SC
                            + (size_t)(ixi - 1) * SX
                            + (size_t)(iyi - 1) * SY
                            + (size_t)(izi - 1);

    float acc0 = 0.f, acc1 = 0.f, acc2 = 0.f, acc3 = 0.f;
#pragma unroll
    for (int x = 0; x < 4; ++x) {
#pragma unroll
        for (int y = 0; y < 4; ++y) {
            const float wxy = wx[x] * wy[y];
            const float* q = base + (size_t)x * SX + (size_t)y * SY;
            const float4 f0 = *(const float4*)(q + 0u * SC);
            const float4 f1 = *(const float4*)(q + 1u * SC);
            const float4 f2 = *(const float4*)(q + 2u * SC);
            const float4 f3 = *(const float4*)(q + 3u * SC);
            acc0 = fmaf(wxy, fmaf(f0.x, wz0, fmaf(f0.y, wz1, fmaf(f0.z, wz2, f0.w * wz3))), acc0);
            acc1 = fmaf(wxy, fmaf(f1.x, wz0, fmaf(f1.y, wz1, fmaf(f1.z, wz2, f1.w * wz3))), acc1);
            acc2 = fmaf(wxy, fmaf(f2.x, wz0, fmaf(f2.y, wz1, fmaf(f2.z, wz2, f2.w * wz3))), acc2);
            acc3 = fmaf(wxy, fmaf(f3.x, wz0, fmaf(f3.y, wz1, fmaf(f3.z, wz2, f3.w * wz3))), acc3);
        }
    }

    if (vbase + vsel < V) {
        float4 o;
        o.x = acc0; o.y = acc1; o.z = acc2; o.w = acc3;
        *(float4*)(out + (size_t)v * NCH + half * 4) = o;   // 16B-aligned
    }
}

extern "C" void kernel_launch(void* const* d_in, const int* in_sizes, int n_in,
                              void* d_out, int out_size, void* d_ws, size_t ws_size,
                              hipStream_t stream) {
    const float* vert = (const float*)d_in[0];   // (1, V, 3) fp32
    const float* vol  = (const float*)d_in[1];   // (1, 8, 112, 224, 160) fp32
    float* out = (float*)d_out;                  // (1, V, 8) fp32

    const int V = in_sizes[0] / 3;
    const int groups = (V + 15) / 16;            // one 16-vertex group per wave
    const int blocks = (groups + 7) / 8;         // 8 waves (256 threads) per block
    cubic_interp_wmma<<<blocks, 256, 0, stream>>>(vert, vol, out, V);
}